// GNN_10986526343434
// MI455X (gfx1250) — compile-verified
//
#include <hip/hip_runtime.h>

typedef float v2f __attribute__((ext_vector_type(2)));
typedef float v8f __attribute__((ext_vector_type(8)));

// ---------------------------------------------------------------------------
// Degree / normalization helpers
// ---------------------------------------------------------------------------
__global__ void gcn_init_deg(float* __restrict__ deg, int n) {
  int i = blockIdx.x * blockDim.x + threadIdx.x;
  if (i < n) deg[i] = 1.0f;  // self-loop contributes 1 to every node's degree
}

__global__ void gcn_count_deg(const int* __restrict__ dst, float* deg, int e) {
  int i = blockIdx.x * blockDim.x + threadIdx.x;
  if (i < e) atomicAdd(&deg[dst[i]], 1.0f);
}

__global__ void gcn_dinv(float* deg, int n) {
  int i = blockIdx.x * blockDim.x + threadIdx.x;
  if (i < n) deg[i] = rsqrtf(deg[i]);  // deg >= 1 always (self-loops)
}

__global__ void gcn_zero(float* __restrict__ p, long n) {
  long i = (long)blockIdx.x * blockDim.x + threadIdx.x;
  if (i < n) p[i] = 0.0f;
}

// ---------------------------------------------------------------------------
// Dense GEMM: C[nrows x LDC] = A[nrows x 64] * B[64 x LDB], f32 WMMA 16x16x4.
// One wave computes one 16x16 C tile; K=64 -> 16 chained WMMAs.
// Compile-time LDB/LDC/NVALID: all B/C addressing folds to base + immediate.
// Columns of B beyond NVALID are *clamped* (not masked): the resulting pad
// columns of C are never read downstream (h2 stride 48, cols 40..47 unused).
// ISA 7.12.2 f32 layouts:
//   A 16x4 : lane l<16 holds {A[l][k],A[l][k+1]}, lane l>=16 -> {k+2,k+3}
//   B 4x16 : VGPR v, lanes 0-15 -> B[k+v][n], lanes 16-31 -> B[k+v+2][n]
//   C 16x16: VGPR r, lanes 0-15 -> row r, lanes 16-31 -> row r+8, col=lane&15
// ---------------------------------------------------------------------------
template <int LDB, int LDC, int NVALID>
__global__ void gcn_gemm_wmma_f32(const float* __restrict__ A,
                                  const float* __restrict__ B,
                                  float* __restrict__ C, int nrows) {
  const int lane = threadIdx.x & 31;
  const int wave = threadIdx.x >> 5;
  const int row0 = (blockIdx.x * (blockDim.x >> 5) + wave) * 16;
  if (row0 >= nrows) return;
  const int n0   = blockIdx.y * 16;

  const int mrow = lane & 15;      // 0..15
  const int hsel = lane >> 4;      // 0 or 1 (lane half)
  const int kofs = hsel << 1;      // 0 or 2

  const float* arow = A + (row0 + mrow) * 64 + kofs;          // 32-bit math
  const int    bn   = n0 + mrow;
  const int    bcol = (bn < NVALID) ? bn : (NVALID - 1);      // branchless clamp
  const float* bp   = B + kofs * LDB + bcol;

  v8f acc = {};
#pragma unroll
  for (int k = 0; k < 64; k += 4) {
    v2f a = *(const v2f*)(arow + k);                 // global_load_b64, imm off
    v2f b = { bp[k * LDB], bp[k * LDB + LDB] };      // 2x b32, imm offsets
    acc = __builtin_amdgcn_wmma_f32_16x16x4_f32(
        /*neg_a=*/false, a, /*neg_b=*/false, b,
        /*c_mod=*/(short)0, acc, /*reuse_a=*/false, /*reuse_b=*/false);
  }

  float* cp = C + (row0 + (hsel << 3)) * LDC + n0 + mrow;
  if (row0 + 16 <= nrows) {        // full tile (always true for N%16==0)
#pragma unroll
    for (int r = 0; r < 8; ++r) cp[r * LDC] = acc[r];
  } else {
    const int crow = row0 + (hsel << 3);
#pragma unroll
    for (int r = 0; r < 8; ++r)
      if (crow + r < nrows) cp[r * LDC] = acc[r];
  }
}

// ---------------------------------------------------------------------------
// Edge scatter, d=64: one wave per edge, float2 per lane (256B coalesced
// gather from L2-resident h; f32 atomics resolve in L2 atomic units).
// ---------------------------------------------------------------------------
__global__ void gcn_scatter64(const float* __restrict__ h,
                              const int* __restrict__ src,
                              const int* __restrict__ dst,
                              const float* __restrict__ dinv,
                              float* agg, int e) {
  int widx = blockIdx.x * (blockDim.x >> 5) + (threadIdx.x >> 5);
  int lane = threadIdx.x & 31;
  if (widx >= e) return;
  int s = src[widx], d = dst[widx];
  float w = dinv[s] * dinv[d];
  v2f v = *(const v2f*)(h + (long)s * 64 + lane * 2);
  float* od = agg + (long)d * 64 + lane * 2;
  atomicAdd(od + 0, v.x * w);
  atomicAdd(od + 1, v.y * w);
}

// Fused: h1 = relu(agg1 + dinv^2 * h (self-loop) + b1), in place over agg1.
__global__ void gcn_epilogue1(float* agg, const float* __restrict__ h,
                              const float* __restrict__ dinv,
                              const float* __restrict__ b1, int n) {
  long i = (long)blockIdx.x * blockDim.x + threadIdx.x;
  if (i >= (long)n * 64) return;
  int node = (int)(i >> 6);
  int c    = (int)(i & 63);
  float di = dinv[node];
  float v  = agg[i] + di * di * h[i] + b1[c];
  agg[i] = v > 0.0f ? v : 0.0f;
}

// Edge scatter, d=40 (h2 padded to stride 48): one wave per edge.
__global__ void gcn_scatter40(const float* __restrict__ h2,
                              const int* __restrict__ src,
                              const int* __restrict__ dst,
                              const float* __restrict__ dinv,
                              float* out, int e) {
  int widx = blockIdx.x * (blockDim.x >> 5) + (threadIdx.x >> 5);
  int lane = threadIdx.x & 31;
  if (widx >= e) return;
  int s = src[widx], d = dst[widx];
  float w = dinv[s] * dinv[d];
  const float* hs = h2 + (long)s * 48;
  float* od = out + (long)d * 40;
  atomicAdd(od + lane, hs[lane] * w);
  if (lane < 8) atomicAdd(od + 32 + lane, hs[32 + lane] * w);
}

// Fused: out += dinv^2 * h2 (self-loop) + b3.
__global__ void gcn_epilogue2(float* out, const float* __restrict__ h2,
                              const float* __restrict__ dinv,
                              const float* __restrict__ b3, int n) {
  long i = (long)blockIdx.x * blockDim.x + threadIdx.x;
  if (i >= (long)n * 40) return;
  int node = (int)(i / 40);
  int c    = (int)(i % 40);
  float di = dinv[node];
  out[i] += di * di * h2[(long)node * 48 + c] + b3[c];
}

// ---------------------------------------------------------------------------
extern "C" void kernel_launch(void* const* d_in, const int* in_sizes, int n_in,
                              void* d_out, int out_size, void* d_ws, size_t ws_size,
                              hipStream_t stream) {
  (void)n_in; (void)out_size; (void)ws_size;
  const float* x  = (const float*)d_in[0];
  const int*   ei = (const int*)d_in[1];     // edge_index [2, E]
  const float* W1 = (const float*)d_in[2];   // [64,64]
  const float* b1 = (const float*)d_in[3];   // [64]
  const float* W3 = (const float*)d_in[4];   // [64,40]
  const float* b3 = (const float*)d_in[5];   // [40]

  const int N = in_sizes[0] / 64;
  const int E = in_sizes[1] / 2;
  const int* src = ei;
  const int* dst = ei + E;

  // Workspace layout (floats): dinv[N] | h[N*64] | agg1/h1[N*64] | h2[N*48]
  float* dinv = (float*)d_ws;
  float* h    = dinv + N;
  float* agg1 = h + (long)N * 64;
  float* h2   = agg1 + (long)N * 64;
  float* out  = (float*)d_out;

  const int T = 256;
  const int mtiles = (N + 15) / 16;

  // --- degree + symmetric normalization ---
  gcn_init_deg<<<(N + T - 1) / T, T, 0, stream>>>(dinv, N);
  gcn_count_deg<<<(E + T - 1) / T, T, 0, stream>>>(dst, dinv, E);
  gcn_dinv<<<(N + T - 1) / T, T, 0, stream>>>(dinv, N);

  // --- layer 1: h = x @ W1 ---
  {
    dim3 grid((mtiles + 7) / 8, 4);  // 8 waves/block, 4 column tiles (64 cols)
    gcn_gemm_wmma_f32<64, 64, 64><<<grid, 256, 0, stream>>>(x, W1, h, N);
  }
  gcn_zero<<<(int)(((long)N * 64 + T - 1) / T), T, 0, stream>>>(agg1, (long)N * 64);
  gcn_scatter64<<<(E + 7) / 8, 256, 0, stream>>>(h, src, dst, dinv, agg1, E);
  gcn_epilogue1<<<(int)(((long)N * 64 + T - 1) / T), T, 0, stream>>>(agg1, h, dinv, b1, N);

  // --- layer 2: h2 = h1 @ W3 (40 cols, padded to stride 48) ---
  {
    dim3 grid((mtiles + 7) / 8, 3);  // 3 column tiles cover 48 >= 40 cols
    gcn_gemm_wmma_f32<40, 48, 40><<<grid, 256, 0, stream>>>(agg1, W3, h2, N);
  }
  gcn_zero<<<(int)(((long)N * 40 + T - 1) / T), T, 0, stream>>>(out, (long)N * 40);
  gcn_scatter40<<<(E + 7) / 8, 256, 0, stream>>>(h2, src, dst, dinv, out, E);
  gcn_epilogue2<<<(int)(((long)N * 40 + T - 1) / T), T, 0, stream>>>(out, h2, dinv, b3, N);
}